// BWItnBlock_49177375539587
// MI455X (gfx1250) — compile-verified
//
#include <hip/hip_runtime.h>
#include <math.h>

// ---------------------------------------------------------------------------
// IterNorm (whitening batch norm) forward, fp32, WMMA f32_16x16x4 path.
// X: [B=128, C=64, H=56, W=56]; d = 64 channels, m = B*H*W = 401408 columns.
// ---------------------------------------------------------------------------

#define D_CH   64
#define HW     3136           // 56*56, = 49 * 64
#define BLKS_PER_B 49         // HW / 64
#define M_COLS 401408         // 128 * 3136
#define NBLOCKS (128 * BLKS_PER_B)   // 6272 column tiles of 64
#define EPSV   1e-05f
#define T_ITER 10
#define LDA    68             // padded LDS row stride (floats): 68 % 64 = 4 -> conflict-free column reads

typedef __attribute__((ext_vector_type(2))) float v2f;
typedef __attribute__((ext_vector_type(8))) float v8f;

__device__ __forceinline__ v8f wmma4(v2f a, v2f b, v8f c) {
    // D = A(16x4) * B(4x16) + C(16x16), all fp32
    return __builtin_amdgcn_wmma_f32_16x16x4_f32(
        /*neg_a=*/false, a, /*neg_b=*/false, b,
        /*c_mod=*/(short)0, c, /*reuse_a=*/false, /*reuse_b=*/false);
}

// ---------------------------------------------------------------------------
// Kernel 0: zero the Gram + sums accumulators in workspace
// ---------------------------------------------------------------------------
__global__ void knl_zero(float* p, int n) {
    int i = blockIdx.x * blockDim.x + threadIdx.x;
    if (i < n) p[i] = 0.0f;
}

// ---------------------------------------------------------------------------
// Kernel 1: Gram accumulation  G[c1][c2] += sum_j x[c1,j]*x[c2,j], plus sums.
// 128 threads = 4 waves. Wave w owns output row band [16w, 16w+16).
// Grid-stride over 64-column tiles; atomics only once at the end.
// ---------------------------------------------------------------------------
__global__ void __launch_bounds__(128)
knl_gram(const float* __restrict__ X, float* __restrict__ G, float* __restrict__ sums) {
    __shared__ float tile[D_CH * LDA];

    const int t    = threadIdx.x;
    const int wave = t >> 5;
    const int lane = t & 31;
    const int ln16 = lane & 15;
    const int khalf = (lane < 16) ? 0 : 2;

    const int c16  = (t & 15) * 4;  // column group for loads
    const int rgrp = t >> 4;        // row group 0..7 for loads

    v8f acc[4];
    for (int nb = 0; nb < 4; ++nb) acc[nb] = (v8f){0,0,0,0,0,0,0,0};
    float psum[8];
    for (int r = 0; r < 8; ++r) psum[r] = 0.0f;

    const int mrow = wave * 16 + ln16;

    for (int blk = blockIdx.x; blk < NBLOCKS; blk += gridDim.x) {
        const int b  = blk / BLKS_PER_B;
        const int p0 = (blk % BLKS_PER_B) * 64;
        const float* base = X + (size_t)b * D_CH * HW + p0;

        // cooperative load: 64 channels x 64 cols -> LDS (float4, fully coalesced)
        for (int it = 0; it < 8; ++it) {
            const int row = rgrp + it * 8;
            const float4 v = *(const float4*)(base + (size_t)row * HW + c16);
            float* drow = &tile[row * LDA + c16];
            drow[0] = v.x; drow[1] = v.y; drow[2] = v.z; drow[3] = v.w;
            psum[it] += v.x + v.y + v.z + v.w;   // per-channel partial sum
        }
        __syncthreads();

        // Gram tiles: A[m][k]=x[mband+m, j0+k], B[k][n]=x[nband+n, j0+k]
        for (int k = 0; k < 64; k += 4) {
            v2f a;
            a.x = tile[mrow * LDA + k + khalf];
            a.y = tile[mrow * LDA + k + khalf + 1];
            for (int nb = 0; nb < 4; ++nb) {
                const int nrow = nb * 16 + ln16;
                v2f bb;
                bb.x = tile[nrow * LDA + k + khalf];
                bb.y = tile[nrow * LDA + k + khalf + 1];
                acc[nb] = wmma4(a, bb, acc[nb]);
            }
        }
        __syncthreads();
    }

    // flush accumulators (C/D layout: VGPR r -> M = r (+8 for upper half-wave))
    const int rbase = wave * 16 + ((lane < 16) ? 0 : 8);
    for (int nb = 0; nb < 4; ++nb) {
        const int col = nb * 16 + ln16;
        for (int r = 0; r < 8; ++r)
            atomicAdd(&G[(rbase + r) * D_CH + col], acc[nb][r]);
    }
    for (int it = 0; it < 8; ++it)
        atomicAdd(&sums[rgrp + it * 8], psum[it]);
}

// ---------------------------------------------------------------------------
// helpers for the single-workgroup Newton-Schulz kernel
// ---------------------------------------------------------------------------
__device__ __forceinline__ v8f mm64_tile(const float* L, const float* R,
                                         int mb, int nb, int lane) {
    v8f c = (v8f){0,0,0,0,0,0,0,0};
    const int ln16 = lane & 15;
    const int khalf = (lane < 16) ? 0 : 2;
    const int mrow = mb * 16 + ln16;
    const int ncol = nb * 16 + ln16;
    for (int k = 0; k < 64; k += 4) {
        v2f a, b;
        a.x = L[mrow * LDA + k + khalf];
        a.y = L[mrow * LDA + k + khalf + 1];
        b.x = R[(k + khalf) * LDA + ncol];
        b.y = R[(k + khalf + 1) * LDA + ncol];
        c = wmma4(a, b, c);
    }
    return c;
}

__device__ __forceinline__ void store_tile(float* Dm, int mb, int nb, int lane, v8f c) {
    const int col = nb * 16 + (lane & 15);
    const int rbase = mb * 16 + ((lane < 16) ? 0 : 8);
    for (int r = 0; r < 8; ++r) Dm[(rbase + r) * LDA + col] = c[r];
}

// ---------------------------------------------------------------------------
// Kernel 2: Sigma assembly + 10 Newton-Schulz iterations (one WG, 8 waves).
//   Sigma = fix(eps*I + G/m - mean*mean^T); S = Sigma / trace(Sigma)
//   P <- 1.5P - 0.5 P^3 S  (x10);  wm = P*sqrt(rTr);  bias = beta - wm*mean
// ---------------------------------------------------------------------------
__global__ void __launch_bounds__(256)
knl_sigma_ns(const float* __restrict__ G, const float* __restrict__ sums,
             const float* __restrict__ beta,
             float* __restrict__ meanOut, float* __restrict__ wmOut,
             float* __restrict__ biasOut) {
    __shared__ float P [D_CH * LDA];
    __shared__ float S [D_CH * LDA];
    __shared__ float Tm[D_CH * LDA];
    __shared__ float meanS[D_CH];
    __shared__ float rTrS;

    const int t = threadIdx.x;
    const float inv_m = 1.0f / (float)M_COLS;

    if (t < D_CH) meanS[t] = sums[t] * inv_m;
    __syncthreads();

    for (int idx = t; idx < D_CH * D_CH; idx += 256) {
        const int i = idx >> 6, j = idx & 63;
        float v = G[idx] * inv_m - meanS[i] * meanS[j];
        if (i == j) v += EPSV; else v *= 0.9f;      // fix_cov
        S[i * LDA + j] = v;
        P[i * LDA + j] = (i == j) ? 1.0f : 0.0f;
    }
    __syncthreads();

    if (t == 0) {
        float tr = 0.0f;
        for (int i = 0; i < D_CH; ++i) tr += S[i * LDA + i];
        rTrS = 1.0f / tr;
    }
    __syncthreads();
    const float rTr = rTrS;

    for (int idx = t; idx < D_CH * D_CH; idx += 256) {
        const int i = idx >> 6, j = idx & 63;
        S[i * LDA + j] *= rTr;
    }
    __syncthreads();

    const int wave = t >> 5, lane = t & 31;
    // 16 output tiles, 8 waves -> 2 tiles per wave (tile = mb*4 + nb)
    for (int it = 0; it < T_ITER; ++it) {
        v8f acc[2];
        // Tm = P @ P
        for (int q = 0; q < 2; ++q) {
            const int tl = wave * 2 + q;
            acc[q] = mm64_tile(P, P, tl >> 2, tl & 3, lane);
        }
        __syncthreads();
        for (int q = 0; q < 2; ++q) {
            const int tl = wave * 2 + q;
            store_tile(Tm, tl >> 2, tl & 3, lane, acc[q]);
        }
        __syncthreads();
        // Tm = Tm @ P   (= P^3)
        for (int q = 0; q < 2; ++q) {
            const int tl = wave * 2 + q;
            acc[q] = mm64_tile(Tm, P, tl >> 2, tl & 3, lane);
        }
        __syncthreads();
        for (int q = 0; q < 2; ++q) {
            const int tl = wave * 2 + q;
            store_tile(Tm, tl >> 2, tl & 3, lane, acc[q]);
        }
        __syncthreads();
        // P = 1.5P - 0.5 * (Tm @ S)
        for (int q = 0; q < 2; ++q) {
            const int tl = wave * 2 + q;
            acc[q] = mm64_tile(Tm, S, tl >> 2, tl & 3, lane);
        }
        __syncthreads();
        for (int q = 0; q < 2; ++q) {
            const int tl = wave * 2 + q;
            const int mb = tl >> 2, nb = tl & 3;
            const int col = nb * 16 + (lane & 15);
            const int rbase = mb * 16 + ((lane < 16) ? 0 : 8);
            for (int r = 0; r < 8; ++r) {
                const float pv = P[(rbase + r) * LDA + col];
                P[(rbase + r) * LDA + col] = 1.5f * pv - 0.5f * acc[q][r];
            }
        }
        __syncthreads();
    }

    const float sq = sqrtf(rTr);
    for (int idx = t; idx < D_CH * D_CH; idx += 256) {
        const int i = idx >> 6, j = idx & 63;
        wmOut[idx] = P[i * LDA + j] * sq;
    }
    if (t < D_CH) meanOut[t] = meanS[t];
    __syncthreads();
    if (t < D_CH) {
        float dot = 0.0f;
        for (int k = 0; k < D_CH; ++k) dot += P[t * LDA + k] * sq * meanS[k];
        biasOut[t] = beta[t] - dot;
    }
}

// ---------------------------------------------------------------------------
// Kernel 3: apply  out[:, j] = wm @ x[:, j] + bias   (bias folds beta & mean)
// 128 threads = 4 waves; wave w owns output channel band [16w, 16w+16) and
// keeps its wm A-fragments (16x64) in registers. One 64-column tile per WG.
// ---------------------------------------------------------------------------
__global__ void __launch_bounds__(128)
knl_apply(const float* __restrict__ X, const float* __restrict__ wm,
          const float* __restrict__ bias, float* __restrict__ out) {
    __shared__ float tile[D_CH * LDA];

    const int t    = threadIdx.x;
    const int wave = t >> 5;
    const int lane = t & 31;
    const int ln16 = lane & 15;
    const int khalf = (lane < 16) ? 0 : 2;
    const int mbase = wave * 16;

    // preload wm band fragments (hits L2; wm is only 16 KB total)
    v2f afr[16];
    const int mrow = mbase + ln16;
    for (int k = 0; k < 16; ++k) {
        afr[k].x = wm[mrow * D_CH + 4 * k + khalf];
        afr[k].y = wm[mrow * D_CH + 4 * k + khalf + 1];
    }
    const int rbase = mbase + ((lane < 16) ? 0 : 8);
    float bv[8];
    for (int r = 0; r < 8; ++r) bv[r] = bias[rbase + r];

    const int blk = blockIdx.x;
    const int b   = blk / BLKS_PER_B;
    const int p0  = (blk % BLKS_PER_B) * 64;
    const float* src = X   + (size_t)b * D_CH * HW + p0;
    float*       dst = out + (size_t)b * D_CH * HW + p0;

    // stage X tile (64 ch x 64 cols) through LDS, coalesced float4
    const int c16  = (t & 15) * 4;
    const int rgrp = t >> 4;
    for (int it = 0; it < 8; ++it) {
        const int row = rgrp + it * 8;
        const float4 v = *(const float4*)(src + (size_t)row * HW + c16);
        float* drow = &tile[row * LDA + c16];
        drow[0] = v.x; drow[1] = v.y; drow[2] = v.z; drow[3] = v.w;
    }
    __syncthreads();

    for (int nt = 0; nt < 4; ++nt) {
        v8f acc = (v8f){0,0,0,0,0,0,0,0};
        const int nc = nt * 16 + ln16;
        for (int k = 0; k < 16; ++k) {
            v2f bb;
            bb.x = tile[(4 * k + khalf) * LDA + nc];
            bb.y = tile[(4 * k + khalf + 1) * LDA + nc];
            acc = wmma4(afr[k], bb, acc);
        }
        for (int r = 0; r < 8; ++r)
            dst[(size_t)(rbase + r) * HW + nc] = acc[r] + bv[r];
    }
}

// ---------------------------------------------------------------------------
// launcher
// ---------------------------------------------------------------------------
extern "C" void kernel_launch(void* const* d_in, const int* in_sizes, int n_in,
                              void* d_out, int out_size, void* d_ws, size_t ws_size,
                              hipStream_t stream) {
    (void)in_sizes; (void)n_in; (void)out_size; (void)ws_size;
    const float* X    = (const float*)d_in[0];
    const float* beta = (const float*)d_in[1];
    float* out = (float*)d_out;
    float* ws  = (float*)d_ws;

    float* G    = ws;                 // 4096 floats
    float* sums = ws + 4096;          //   64
    float* mean = ws + 4160;          //   64
    float* wm   = ws + 4224;          // 4096
    float* bias = ws + 8320;          //   64

    knl_zero<<<(4160 + 255) / 256, 256, 0, stream>>>(G, 4160);
    knl_gram<<<896, 128, 0, stream>>>(X, G, sums);
    knl_sigma_ns<<<1, 256, 0, stream>>>(G, sums, beta, mean, wm, bias);
    knl_apply<<<NBLOCKS, 128, 0, stream>>>(X, wm, bias, out);
}